// SURVEY_BA_2grid_Cheb_21930103013657
// MI455X (gfx1250) — compile-verified
//
#include <hip/hip_runtime.h>

// ---------------------------------------------------------------------------
// ChebConv GNN for MI455X (gfx1250). FP32 throughout (workload is L2/atomic
// bound, not matmul bound). Dense node-feature GEMMs use V_WMMA_F32_16X16X4_F32.
// ---------------------------------------------------------------------------

typedef __attribute__((ext_vector_type(2))) float v2f;
typedef __attribute__((ext_vector_type(8))) float v8f;

#define HID 30

// ---------------- elementwise / setup kernels ------------------------------

__global__ void zero_kernel(float* __restrict__ p, int n) {
  int i = blockIdx.x * blockDim.x + threadIdx.x;
  if (i < n) p[i] = 0.0f;
}

__global__ void deg_kernel(const int* __restrict__ rows, const int* __restrict__ cols,
                           const float* __restrict__ ew, float* __restrict__ deg, int ne) {
  int e = blockIdx.x * blockDim.x + threadIdx.x;
  if (e >= ne) return;
  int r = rows[e];
  float w = (r == cols[e]) ? 0.0f : ew[e];
  if (w != 0.0f) atomicAdd(&deg[r], w);
}

__global__ void dinv_kernel(const float* __restrict__ deg, float* __restrict__ dinv, int n) {
  int i = blockIdx.x * blockDim.x + threadIdx.x;
  if (i >= n) return;
  float d = deg[i];
  dinv[i] = (d > 0.0f) ? rsqrtf(fmaxf(d, 1e-30f)) : 0.0f;
}

__global__ void norm_kernel(const int* __restrict__ rows, const int* __restrict__ cols,
                            const float* __restrict__ ew, const float* __restrict__ dinv,
                            float* __restrict__ nrm, int ne) {
  int e = blockIdx.x * blockDim.x + threadIdx.x;
  if (e >= ne) return;
  int r = rows[e], c = cols[e];
  float w = (r == c) ? 0.0f : ew[e];
  nrm[e] = -dinv[r] * w * dinv[c];
}

// out[col[e], :] += norm[e] * x[row[e], :]   (gather + scatter-add; all in L2)
// Row bases are 8-byte aligned (F even), so gather as float2.
template <int F>
__global__ void spmv_kernel(const int* __restrict__ rows, const int* __restrict__ cols,
                            const float* __restrict__ nrm, const float* __restrict__ X,
                            float* __restrict__ Y, int ne) {
  int e = blockIdx.x * blockDim.x + threadIdx.x;
  if (e >= ne) return;
  float nv = nrm[e];
  if (nv == 0.0f) return;
  const float2* __restrict__ xr = (const float2*)(X + (size_t)rows[e] * F);
  float* __restrict__ yc = Y + (size_t)cols[e] * F;
  float2 v[F / 2];
#pragma unroll
  for (int f = 0; f < F / 2; ++f) v[f] = xr[f];
#pragma unroll
  for (int f = 0; f < F / 2; ++f) {
    atomicAdd(yc + 2 * f,     nv * v[f].x);
    atomicAdd(yc + 2 * f + 1, nv * v[f].y);
  }
}

// Chebyshev recurrence tail: C = 2*C - A
__global__ void recur_kernel(float* __restrict__ C, const float* __restrict__ A, int n) {
  int i = blockIdx.x * blockDim.x + threadIdx.x;
  if (i < n) C[i] = 2.0f * C[i] - A[i];
}

__global__ void bias_relu_kernel(const float* __restrict__ S, const float* __restrict__ b,
                                 float* __restrict__ H, int total, int F) {
  int i = blockIdx.x * blockDim.x + threadIdx.x;
  if (i >= total) return;
  float v = S[i] + b[i % F];
  H[i] = v > 0.0f ? v : 0.0f;
}

// ---------------- WMMA GEMM: OUT(nrows x 30) += T(nrows x F) @ W(F x 30) ----
// One wave32 per 16-row tile; K in steps of 4 via V_WMMA_F32_16X16X4_F32.
// A layout (16x4 f32):  lane m = lane&15, k0 = 2*(lane>>4); a = {A[m][k0], A[m][k0+1]}
// B layout (4x16 f32):  lane n = lane&15, k0 = 2*(lane>>4); b = {B[k0][n], B[k0+1][n]}
// C/D layout (16x16):   vgpr r holds D[r + 8*(lane>>4)][lane&15]
//
// Codegen strategy: no predicated loads (addresses clamped, only B K-pad is
// value-zeroed) and a branch-free atomic-add epilogue: padded lanes add 0.0f
// to a clamped-but-valid address, which is race-free under atomics.

template <int F>
__global__ void gemm_cheb_wmma(const float* __restrict__ T, const float* __restrict__ W,
                               float* __restrict__ OUT, int nrows) {
  constexpr int HOUT = HID;            // 30
  constexpr int KC   = (F + 3) / 4;    // K chunks of 4

  const int wave = threadIdx.x >> 5;
  const int lane = threadIdx.x & 31;
  const int tile = (blockIdx.x * (blockDim.x >> 5) + wave) << 4;
  if (tile >= nrows) return;           // uniform per wave: EXEC all-ones for WMMA

  const int m    = lane & 15;
  const int half = lane >> 4;
  const int arow = min(tile + m, nrows - 1);          // clamped (discarded rows)
  const int n1   = min(m + 16, HOUT - 1);             // clamped (padded cols add 0)
  const bool colok = (m + 16 < HOUT);
  const float* __restrict__ Trow = T + (size_t)arow * F;

  // Preload B fragments for both column tiles (W is tiny, lives in L2).
  v2f bf0[KC], bf1[KC];
#pragma unroll
  for (int kc = 0; kc < KC; ++kc) {
    const int  k0 = kc * 4 + half * 2;
    const bool v0 = (k0 < F), v1 = (k0 + 1 < F);      // compile-time except boundary chunk
    const int  i0 = v0 ? k0 : 0;
    const int  i1 = v1 ? (k0 + 1) : 0;
    bf0[kc].x = v0 ? W[i0 * HOUT + m]  : 0.0f;
    bf0[kc].y = v1 ? W[i1 * HOUT + m]  : 0.0f;
    bf1[kc].x = v0 ? W[i0 * HOUT + n1] : 0.0f;
    bf1[kc].y = v1 ? W[i1 * HOUT + n1] : 0.0f;
  }

  v8f acc0 = {};  // cols 0..15
  v8f acc1 = {};  // cols 16..31 (cols >= 30 padded, contribute 0)

#pragma unroll
  for (int kc = 0; kc < KC; ++kc) {
    const int k0 = kc * 4 + half * 2;
    const int i0 = (k0 < F) ? k0 : 0;                 // clamped; B is zero there
    const int i1 = (k0 + 1 < F) ? (k0 + 1) : 0;
    v2f a;
    a.x = Trow[i0];
    a.y = Trow[i1];
    acc0 = __builtin_amdgcn_wmma_f32_16x16x4_f32(false, a, false, bf0[kc], (short)0, acc0, false, false);
    acc1 = __builtin_amdgcn_wmma_f32_16x16x4_f32(false, a, false, bf1[kc], (short)0, acc1, false, false);
  }

  // Branch-free epilogue: atomic adds; padded lanes contribute exactly 0.0f.
  const int mr0 = tile + half * 8;
  if (tile + 16 <= nrows) {            // uniform: full tile (all but the last one)
    float* __restrict__ o = OUT + (size_t)mr0 * HOUT;
#pragma unroll
    for (int r = 0; r < 8; ++r) {
      atomicAdd(o + r * HOUT + m,  acc0[r]);
      atomicAdd(o + r * HOUT + n1, colok ? acc1[r] : 0.0f);
    }
  } else {                             // last partial tile: clamp rows, zero values
#pragma unroll
    for (int r = 0; r < 8; ++r) {
      const int  mr  = mr0 + r;
      const bool rok = mr < nrows;
      float* __restrict__ o = OUT + (size_t)min(mr, nrows - 1) * HOUT;
      atomicAdd(o + m,  rok ? acc0[r] : 0.0f);
      atomicAdd(o + n1, (rok && colok) ? acc1[r] : 0.0f);
    }
  }
}

// ---------------- pooling + MLP --------------------------------------------

__global__ void pool_kernel(const float* __restrict__ H, const int* __restrict__ batch,
                            float* __restrict__ sums, float* __restrict__ cnt, int n, int F) {
  int i = blockIdx.x * blockDim.x + threadIdx.x;
  if (i >= n) return;
  int g = batch[i];
  atomicAdd(&cnt[g], 1.0f);
  const float* __restrict__ hr = H + (size_t)i * F;
  float* __restrict__ sg = sums + (size_t)g * F;
  for (int f = 0; f < F; ++f) atomicAdd(sg + f, hr[f]);
}

__global__ void mlp_kernel(const float* __restrict__ sums, const float* __restrict__ cnt,
                           const float* __restrict__ l1w, const float* __restrict__ l1b,
                           const float* __restrict__ l2w, const float* __restrict__ l2b,
                           float* __restrict__ out, int ngraphs) {
  int g = blockIdx.x * blockDim.x + threadIdx.x;
  if (g >= ngraphs) return;
  float inv = 1.0f / fmaxf(cnt[g], 1.0f);
  float p[HID];
#pragma unroll
  for (int f = 0; f < HID; ++f) p[f] = sums[(size_t)g * HID + f] * inv;
  float h1[HID];
  for (int j = 0; j < HID; ++j) {
    float s = l1b[j];
#pragma unroll
    for (int f = 0; f < HID; ++f) s += p[f] * l1w[f * HID + j];
    h1[j] = s > 0.0f ? s : 0.0f;
  }
  for (int c = 0; c < 2; ++c) {
    float s = l2b[c];
#pragma unroll
    for (int j = 0; j < HID; ++j) s += h1[j] * l2w[j * 2 + c];
    out[g * 2 + c] = s;
  }
}

// ---------------- host orchestration ---------------------------------------

extern "C" void kernel_launch(void* const* d_in, const int* in_sizes, int n_in,
                              void* d_out, int out_size, void* d_ws, size_t ws_size,
                              hipStream_t stream) {
  const float* x   = (const float*)d_in[0];
  const int*   ei  = (const int*)d_in[1];
  const int*   bat = (const int*)d_in[2];
  const float* ew  = (const float*)d_in[3];
  const float* W1  = (const float*)d_in[4];
  const float* b1  = (const float*)d_in[5];
  const float* W2  = (const float*)d_in[6];
  const float* b2  = (const float*)d_in[7];
  const float* l1w = (const float*)d_in[8];
  const float* l1b = (const float*)d_in[9];
  const float* l2w = (const float*)d_in[10];
  const float* l2b = (const float*)d_in[11];

  const int F_IN = 10;
  const int nN = in_sizes[0] / F_IN;      // 150000
  const int nE = in_sizes[3];             // 2400000
  const int nG = out_size / 2;            // 128

  const int* rows = ei;
  const int* cols = ei + nE;

  // workspace layout (floats)
  float* ws   = (float*)d_ws;
  float* deg  = ws;
  float* dinv = deg  + nN;
  float* nrm  = dinv + nN;
  float* TA   = nrm  + nE;
  float* TB   = TA   + (size_t)nN * HID;
  float* TC   = TB   + (size_t)nN * HID;
  float* OUT  = TC   + (size_t)nN * HID;
  float* POOL = OUT  + (size_t)nN * HID;
  float* CNT  = POOL + (size_t)nG * HID;

  const int B = 256;
  auto grid = [&](int n) { return (n + B - 1) / B; };
  auto zero = [&](float* p, int n) { zero_kernel<<<grid(n), B, 0, stream>>>(p, n); };
  const int gemm_tiles  = (nN + 15) / 16;
  const int gemm_blocks = (gemm_tiles + 7) / 8;  // 8 waves (one 16-row tile each) / block
  auto gemm10 = [&](const float* T, const float* Wk) {
    gemm_cheb_wmma<10><<<gemm_blocks, B, 0, stream>>>(T, Wk, OUT, nN);
  };
  auto gemm30 = [&](const float* T, const float* Wk) {
    gemm_cheb_wmma<30><<<gemm_blocks, B, 0, stream>>>(T, Wk, OUT, nN);
  };

  // edge normalization
  zero(deg, nN);
  deg_kernel<<<grid(nE), B, 0, stream>>>(rows, cols, ew, deg, nE);
  dinv_kernel<<<grid(nN), B, 0, stream>>>(deg, dinv, nN);
  norm_kernel<<<grid(nE), B, 0, stream>>>(rows, cols, ew, dinv, nrm, nE);

  // ---------------- layer 1: F=10 -> HID, W1[k] is 10x30 ----------------
  {
    const int F = 10, WS = F * HID;
    zero(OUT, nN * HID);
    gemm10(x, W1 + 0 * WS);                                         // T0 = x
    zero(TB, nN * F);
    spmv_kernel<10><<<grid(nE), B, 0, stream>>>(rows, cols, nrm, x, TB, nE);   // T1
    gemm10(TB, W1 + 1 * WS);
    zero(TC, nN * F);
    spmv_kernel<10><<<grid(nE), B, 0, stream>>>(rows, cols, nrm, TB, TC, nE);
    recur_kernel<<<grid(nN * F), B, 0, stream>>>(TC, x, nN * F);    // T2 = 2L T1 - T0
    gemm10(TC, W1 + 2 * WS);
    zero(TA, nN * F);
    spmv_kernel<10><<<grid(nE), B, 0, stream>>>(rows, cols, nrm, TC, TA, nE);
    recur_kernel<<<grid(nN * F), B, 0, stream>>>(TA, TB, nN * F);   // T3
    gemm10(TA, W1 + 3 * WS);
    zero(TB, nN * F);
    spmv_kernel<10><<<grid(nE), B, 0, stream>>>(rows, cols, nrm, TA, TB, nE);
    recur_kernel<<<grid(nN * F), B, 0, stream>>>(TB, TC, nN * F);   // T4
    gemm10(TB, W1 + 4 * WS);
    // H1 = relu(OUT + b1) -> TA (stride 30)
    bias_relu_kernel<<<grid(nN * HID), B, 0, stream>>>(OUT, b1, TA, nN * HID, HID);
  }

  // ---------------- layer 2: F=30 -> HID, W2[k] is 30x30 ----------------
  {
    const int F = HID, WS = F * HID;
    zero(OUT, nN * HID);
    gemm30(TA, W2 + 0 * WS);                                        // T0 = H1 (in TA)
    zero(TB, nN * F);
    spmv_kernel<30><<<grid(nE), B, 0, stream>>>(rows, cols, nrm, TA, TB, nE);  // T1
    gemm30(TB, W2 + 1 * WS);
    zero(TC, nN * F);
    spmv_kernel<30><<<grid(nE), B, 0, stream>>>(rows, cols, nrm, TB, TC, nE);
    recur_kernel<<<grid(nN * F), B, 0, stream>>>(TC, TA, nN * F);   // T2
    gemm30(TC, W2 + 2 * WS);
    zero(TA, nN * F);                                               // H1 no longer needed
    spmv_kernel<30><<<grid(nE), B, 0, stream>>>(rows, cols, nrm, TC, TA, nE);
    recur_kernel<<<grid(nN * F), B, 0, stream>>>(TA, TB, nN * F);   // T3
    gemm30(TA, W2 + 3 * WS);
    zero(TB, nN * F);
    spmv_kernel<30><<<grid(nE), B, 0, stream>>>(rows, cols, nrm, TA, TB, nE);
    recur_kernel<<<grid(nN * F), B, 0, stream>>>(TB, TC, nN * F);   // T4
    gemm30(TB, W2 + 4 * WS);
    // H2 = relu(OUT + b2), in place
    bias_relu_kernel<<<grid(nN * HID), B, 0, stream>>>(OUT, b2, OUT, nN * HID, HID);
  }

  // ---------------- pool + MLP ----------------
  zero(POOL, nG * HID);
  zero(CNT, nG);
  pool_kernel<<<grid(nN), B, 0, stream>>>(OUT, bat, POOL, CNT, nN, HID);
  mlp_kernel<<<1, 128, 0, stream>>>(POOL, CNT, l1w, l1b, l2w, l2b, (float*)d_out, nG);
}